// Topkpool_20641612825047
// MI455X (gfx1250) — compile-verified
//
#include <hip/hip_runtime.h>
#include <hip/hip_bf16.h>
#include <math.h>

#define D 128
#define LDA 132      // padded LDS row stride (floats) -> conflict-free column reads
#define EPS 1e-5f

typedef __attribute__((ext_vector_type(2))) float v2f;
typedef __attribute__((ext_vector_type(8))) float v8f;

// ---------------- utility ----------------
__global__ void zero_f32(float* __restrict__ p, size_t n) {
  size_t i = (size_t)blockIdx.x * blockDim.x + threadIdx.x;
  if (i < n) p[i] = 0.0f;
}

// ---------------- degree / normalization ----------------
__global__ void deg_count(const int* __restrict__ ei, int E,
                          float* __restrict__ degO, float* __restrict__ degI) {
  int e = blockIdx.x * blockDim.x + threadIdx.x;
  if (e < E) {
    atomicAdd(&degO[ei[e]],     1.0f);   // src degrees
    atomicAdd(&degI[ei[E + e]], 1.0f);   // dst degrees
  }
}

__global__ void deg_to_inv(float* __restrict__ d, int n) {
  int i = blockIdx.x * blockDim.x + threadIdx.x;
  if (i < n) d[i] = rsqrtf(fmaxf(d[i], 1.0f));
}

// ---------------- SpMM scatter: agg[dst] += h[src] * inv_out[src] ----------------
// One wave per edge: coalesced 512B row gather, 4 float atomics per lane.
__global__ __launch_bounds__(256) void gcn_scatter(
    const float* __restrict__ h, const int* __restrict__ ei,
    const float* __restrict__ invO, float* __restrict__ agg, int E) {
  int e = blockIdx.x * 8 + (threadIdx.x >> 5);
  if (e >= E) return;
  int lane = threadIdx.x & 31;
  int src = ei[e];
  int dst = ei[E + e];
  float s = invO[src];
  float4 v = ((const float4*)(h + (size_t)src * D))[lane];
  float* p = agg + (size_t)dst * D + lane * 4;
  atomicAdd(p + 0, v.x * s);
  atomicAdd(p + 1, v.y * s);
  atomicAdd(p + 2, v.z * s);
  atomicAdd(p + 3, v.w * s);
}

// ---------------- GEMM: Y = (A * inv_in) @ W + bias, in place (Y == A) ----------------
// Block = 256 threads = 8 waves; block owns 16 output rows, wave w owns cols [16w,16w+16).
// A tile staged to LDS (inv_in fused); K swept 4 at a time with V_WMMA_F32_16X16X4_F32.
// Uniform full-tile fast path keeps the epilogue branch-free per lane.
__global__ __launch_bounds__(256) void gcn_gemm_wmma(
    float* __restrict__ A,               // agg in, Y out (same buffer)
    const float* __restrict__ invin,
    const float* __restrict__ W,         // [K=128][N=128] row-major
    const float* __restrict__ bias,
    int Nn) {
  __shared__ float At[16 * LDA];
  const int row0 = blockIdx.x * 16;
  const int t = threadIdx.x;
  const bool fullTile = (row0 + 16 <= Nn);   // uniform across block

  // cooperative staged load of the 16x128 A tile, scaled by inv_in[row]
  if (fullTile) {
    for (int i = t; i < 16 * (D / 4); i += 256) {
      int r  = i >> 5;       // row in tile
      int c4 = i & 31;       // float4 column
      int row = row0 + r;
      float4 v = ((const float4*)(A + (size_t)row * D))[c4];
      float s = invin[row];
      float* dstp = &At[r * LDA + c4 * 4];
      dstp[0] = v.x * s; dstp[1] = v.y * s; dstp[2] = v.z * s; dstp[3] = v.w * s;
    }
  } else {
    for (int i = t; i < 16 * (D / 4); i += 256) {
      int r  = i >> 5;
      int c4 = i & 31;
      int row = row0 + r;
      float4 v = make_float4(0.f, 0.f, 0.f, 0.f);
      float s = 0.f;
      if (row < Nn) {
        v = ((const float4*)(A + (size_t)row * D))[c4];
        s = invin[row];
      }
      float* dstp = &At[r * LDA + c4 * 4];
      dstp[0] = v.x * s; dstp[1] = v.y * s; dstp[2] = v.z * s; dstp[3] = v.w * s;
    }
  }
  __syncthreads();

  const int wave = t >> 5;
  const int lane = t & 31;
  const int hi   = lane >> 4;    // 0: lanes 0-15, 1: lanes 16-31
  const int lrow = lane & 15;
  const int col  = wave * 16 + lrow;

  v8f c = {};
  #pragma unroll 4
  for (int k = 0; k < D; k += 4) {
    // A frag 16x4 fp32: lanes 0-15 -> K=k,k+1 ; lanes 16-31 -> K=k+2,k+3 (M = lane%16)
    v2f a, b;
    a.x = At[lrow * LDA + k + 2 * hi];
    a.y = At[lrow * LDA + k + 2 * hi + 1];
    // B frag 4x16 fp32: mirrored over N (N = lane%16)
    b.x = W[(size_t)(k + 2 * hi) * D + col];
    b.y = W[(size_t)(k + 2 * hi + 1) * D + col];
    c = __builtin_amdgcn_wmma_f32_16x16x4_f32(false, a, false, b, (short)0, c,
                                              false, false);
  }

  const float bb = bias[col];
  if (fullTile) {
    // C/D layout: VGPR g -> M=g (lanes 0-15) / M=g+8 (lanes 16-31)
    float* base = A + (size_t)(row0 + 8 * hi) * D + col;
    #pragma unroll
    for (int g = 0; g < 8; ++g) base[(size_t)g * D] = c[g] + bb;
  } else {
    #pragma unroll
    for (int g = 0; g < 8; ++g) {
      int row = row0 + g + 8 * hi;
      if (row < Nn) A[(size_t)row * D + col] = c[g] + bb;
    }
  }
}

// ---------------- BN stats: per-column sum / sumsq ----------------
__global__ __launch_bounds__(128) void bn_stats(
    const float* __restrict__ y, float* __restrict__ gsum,
    float* __restrict__ gsq, int Nn) {
  int col = threadIdx.x;   // 0..127
  float s = 0.f, q = 0.f;
  for (int r = blockIdx.x; r < Nn; r += gridDim.x) {
    float v = y[(size_t)r * D + col];
    s += v;
    q += v * v;
  }
  atomicAdd(&gsum[col], s);
  atomicAdd(&gsq[col],  q);
}

__global__ __launch_bounds__(128) void bn_finalize(
    const float* __restrict__ gsum, const float* __restrict__ gsq,
    const float* __restrict__ gamma, const float* __restrict__ beta,
    float* __restrict__ scl, float* __restrict__ shf, int Nn) {
  int c = threadIdx.x;
  float invN = 1.0f / (float)Nn;
  float mu  = gsum[c] * invN;
  float var = gsq[c] * invN - mu * mu;
  float sc  = gamma[c] * rsqrtf(var + EPS);
  scl[c] = sc;
  shf[c] = beta[c] - mu * sc;
}

// ---------------- BN apply + ReLU (vectorized) ----------------
__global__ __launch_bounds__(256) void bn_relu(
    const float* __restrict__ y, const float* __restrict__ scl,
    const float* __restrict__ shf, float* __restrict__ out, size_t total4) {
  size_t i = (size_t)blockIdx.x * blockDim.x + threadIdx.x;
  if (i >= total4) return;
  int c = ((int)(i & 31)) * 4;           // 32 float4 per 128-wide row
  float4 v = ((const float4*)y)[i];
  float4 o;
  o.x = fmaxf(v.x * scl[c + 0] + shf[c + 0], 0.f);
  o.y = fmaxf(v.y * scl[c + 1] + shf[c + 1], 0.f);
  o.z = fmaxf(v.z * scl[c + 2] + shf[c + 2], 0.f);
  o.w = fmaxf(v.w * scl[c + 3] + shf[c + 3], 0.f);
  ((float4*)out)[i] = o;
}

// ---------------- host-side orchestration ----------------
extern "C" void kernel_launch(void* const* d_in, const int* in_sizes, int n_in,
                              void* d_out, int out_size, void* d_ws, size_t ws_size,
                              hipStream_t stream) {
  const float* x  = (const float*)d_in[0];
  const int*   ei = (const int*)d_in[1];
  const float* Wm[3] = {(const float*)d_in[2], (const float*)d_in[6],  (const float*)d_in[10]};
  const float* bm[3] = {(const float*)d_in[3], (const float*)d_in[7],  (const float*)d_in[11]};
  const float* gm[3] = {(const float*)d_in[4], (const float*)d_in[8],  (const float*)d_in[12]};
  const float* em[3] = {(const float*)d_in[5], (const float*)d_in[9],  (const float*)d_in[13]};

  const int Nn = in_sizes[0] / D;
  const int E  = in_sizes[1] / 2;

  float* ws   = (float*)d_ws;
  float* invO = ws;                                   // [N]
  float* invI = ws + Nn;                              // [N]
  float* agg  = ws + 2 * (size_t)Nn;                  // [N*D] (also Y, in-place)
  float* hbuf = agg + (size_t)Nn * D;                 // [N*D]
  float* gsum = hbuf + (size_t)Nn * D;                // [128]
  float* gsq  = gsum + D;                             // [128]
  float* scl  = gsq + D;                              // [128]
  float* shf  = scl + D;                              // [128]

  const size_t ND = (size_t)Nn * D;

  // degrees -> inv-sqrt normalizers (in place)
  zero_f32<<<(2 * Nn + 255) / 256, 256, 0, stream>>>(invO, (size_t)2 * Nn);
  deg_count<<<(E + 255) / 256, 256, 0, stream>>>(ei, E, invO, invI);
  deg_to_inv<<<(2 * Nn + 255) / 256, 256, 0, stream>>>(invO, 2 * Nn);

  const float* hin = x;
  for (int l = 0; l < 3; ++l) {
    // agg = 0; agg[dst] += h[src] * inv_out[src]
    zero_f32<<<(int)((ND + 255) / 256), 256, 0, stream>>>(agg, ND);
    gcn_scatter<<<(E + 7) / 8, 256, 0, stream>>>(hin, ei, invO, agg, E);

    // Y = (agg * inv_in) @ W + b   (WMMA, in place)
    gcn_gemm_wmma<<<(Nn + 15) / 16, 256, 0, stream>>>(agg, invI, Wm[l], bm[l], Nn);

    // batchnorm stats + finalize
    zero_f32<<<1, 256, 0, stream>>>(gsum, 2 * D);
    bn_stats<<<512, 128, 0, stream>>>(agg, gsum, gsq, Nn);
    bn_finalize<<<1, 128, 0, stream>>>(gsum, gsq, gm[l], em[l], scl, shf, Nn);

    // h = relu(Y * scale + shift)
    float* outp = (l == 2) ? (float*)d_out : hbuf;
    bn_relu<<<(int)((ND / 4 + 255) / 256), 256, 0, stream>>>(agg, scl, shf, outp, ND / 4);
    hin = hbuf;
  }
}